// Model_738734375234
// MI455X (gfx1250) — compile-verified
//
#include <hip/hip_runtime.h>

#define NS 16384   // B*T samples

typedef float v2f __attribute__((ext_vector_type(2)));
typedef float v8f __attribute__((ext_vector_type(8)));

// ---------------------------------------------------------------------------
// Elementwise / conv / reduction kernels (VALU; cheap vs the WMMA GEMMs)
// ---------------------------------------------------------------------------

__global__ void zero_k(float* __restrict__ p, int n) {
  int i = blockIdx.x * blockDim.x + threadIdx.x;
  if (i < n) p[i] = 0.0f;
}

// 1->C conv, kernel 3, pad 1 along width W. y[n][c*W+t]
__global__ void conv3_1toC_k(const float* __restrict__ x, const float* __restrict__ w,
                             const float* __restrict__ bias, float* __restrict__ y,
                             int N, int W, int C) {
  int i = blockIdx.x * blockDim.x + threadIdx.x;
  int total = N * C * W;
  if (i >= total) return;
  int t = i % W;
  int c = (i / W) % C;
  int n = i / (W * C);
  const float* xr = x + (long)n * W;
  const float* wc = w + c * 3;
  float acc = bias ? bias[c] : 0.0f;
  if (t - 1 >= 0) acc += wc[0] * xr[t - 1];
  acc += wc[1] * xr[t];
  if (t + 1 < W)  acc += wc[2] * xr[t + 1];
  y[i] = acc;
}

// Cin->Cout conv, kernel 3, pad 1. x[n][c*W+t], w[o][c][k], y[n][o*W+t]
__global__ void conv3_CtoC_k(const float* __restrict__ x, const float* __restrict__ w,
                             const float* __restrict__ bias, float* __restrict__ y,
                             int N, int W, int Cin, int Cout) {
  int i = blockIdx.x * blockDim.x + threadIdx.x;
  int total = N * Cout * W;
  if (i >= total) return;
  int t = i % W;
  int o = (i / W) % Cout;
  int n = i / (W * Cout);
  const float* xr = x + (long)n * Cin * W;
  const float* wr = w + (long)o * Cin * 3;
  float acc = bias ? bias[o] : 0.0f;
  for (int c = 0; c < Cin; ++c) {
    const float* xc = xr + c * W;
    float s = wr[c * 3 + 1] * xc[t];
    if (t > 0)     s += wr[c * 3 + 0] * xc[t - 1];
    if (t + 1 < W) s += wr[c * 3 + 2] * xc[t + 1];
    acc += s;
  }
  y[i] = acc;
}

// LocallyConnected (kernel=1): y[n][o*W+t] = sum_c x[n][c*W+t] * w[o][c][t]
__global__ void lc_k(const float* __restrict__ x, const float* __restrict__ w,
                     float* __restrict__ y, int N, int W, int Cin, int Cout) {
  int i = blockIdx.x * blockDim.x + threadIdx.x;
  int total = N * Cout * W;
  if (i >= total) return;
  int t = i % W;
  int o = (i / W) % Cout;
  int n = i / (W * Cout);
  const float* xr = x + (long)n * Cin * W;
  const float* wr = w + ((long)o * Cin) * W;
  float acc = 0.0f;
  for (int c = 0; c < Cin; ++c) acc += xr[c * W + t] * wr[c * W + t];
  y[i] = acc;
}

// Per-channel sum / sumsq over (N, C, W) layout [n][c*W+t].  stats[0..C)=sum, [C..2C)=sumsq
__global__ void stats_k(const float* __restrict__ x, float* __restrict__ stats,
                        long total, int C, int W, int reluFirst) {
  __shared__ float ssum[512];
  __shared__ float ssq[512];
  for (int c = threadIdx.x; c < C; c += blockDim.x) { ssum[c] = 0.f; ssq[c] = 0.f; }
  __syncthreads();
  long stride = (long)gridDim.x * blockDim.x;
  for (long i = (long)blockIdx.x * blockDim.x + threadIdx.x; i < total; i += stride) {
    float v = x[i];
    if (reluFirst) v = fmaxf(v, 0.0f);
    int c = (int)((i / W) % C);
    atomicAdd(&ssum[c], v);
    atomicAdd(&ssq[c], v * v);
  }
  __syncthreads();
  for (int c = threadIdx.x; c < C; c += blockDim.x) {
    atomicAdd(&stats[c], ssum[c]);
    atomicAdd(&stats[C + c], ssq[c]);
  }
}

// stats -> scale/shift in place:  scale = g*rsqrt(var+eps), shift = b - mean*scale
__global__ void bnfin_k(float* __restrict__ stats, const float* __restrict__ g,
                        const float* __restrict__ b, int C, float invCnt) {
  int c = blockIdx.x * blockDim.x + threadIdx.x;
  if (c >= C) return;
  float m = stats[c] * invCnt;
  float v = stats[C + c] * invCnt - m * m;
  float sc = g[c] * rsqrtf(v + 1e-5f);
  stats[c]     = sc;
  stats[C + c] = b[c] - m * sc;
}

// reluAfter=1 (EMG): y = relu(x*scale+shift). reluAfter=0 (GLOVE): y = relu(x)*scale+shift
__global__ void bnapply_k(float* __restrict__ x, const float* __restrict__ stats,
                          long total, int C, int W, int reluAfter) {
  long i = (long)blockIdx.x * blockDim.x + threadIdx.x;
  if (i >= total) return;
  int c = (int)((i / W) % C);
  float v = x[i];
  if (reluAfter) v = fmaxf(v * stats[c] + stats[C + c], 0.0f);
  else           v = fmaxf(v, 0.0f) * stats[c] + stats[C + c];
  x[i] = v;
}

// one 128-thread block per row; L2-normalize row of length D (=128)
__global__ void l2norm_k(float* __restrict__ x, int D) {
  __shared__ float red[128];
  long n = blockIdx.x;
  float v = x[n * D + threadIdx.x];
  red[threadIdx.x] = v * v;
  __syncthreads();
  for (int s = 64; s > 0; s >>= 1) {
    if ((int)threadIdx.x < s) red[threadIdx.x] += red[threadIdx.x + s];
    __syncthreads();
  }
  x[n * D + threadIdx.x] = v * rsqrtf(red[0]);
}

// ---------------------------------------------------------------------------
// WMMA fp32 GEMM:  Y = X @ W^T  (+bias)  [* scale],   optionally batched.
//   X: (M,K) row-major, W: (Nn,K) row-major, Y: (M,Nn) row-major.
//   128 threads = 4 wave32s; each wave computes a 32x64 tile via
//   V_WMMA_F32_16X16X4_F32 (2 A-frags x 4 B-frags per K-step of 4).
//   Fragment layouts per CDNA5 ISA 7.12.2:
//     A 16x4 f32: lane(l<16) -> row l, K={0,1}; lane(l>=16) -> row l-16, K={2,3}
//     B 4x16 f32: lane -> column, same K split  => B-column n == W-row n: contiguous.
//     C/D: VGPR v holds row 16i + v + 8*half, col = lane&15.
// ---------------------------------------------------------------------------
__global__ __launch_bounds__(128) void gemm_wmma_k(
    const float* __restrict__ X, const float* __restrict__ Wt,
    const float* __restrict__ bias, float* __restrict__ Y,
    int M, int Nn, int K, long sx, long sw, long sy,
    const float* __restrict__ scale_ptr, float scale_const) {
  int b = blockIdx.z;
  const float* Xb = X + (long)b * sx;
  const float* Wb = Wt + (long)b * sw;
  float* Yb = Y + (long)b * sy;

  int wave = threadIdx.x >> 5;
  int lane = threadIdx.x & 31;
  int half = lane >> 4;
  int idx  = lane & 15;
  int wm = wave >> 1, wn = wave & 1;
  int m0 = (blockIdx.y * 2 + wm) * 32;
  int n0 = (blockIdx.x * 2 + wn) * 64;

  v8f acc[2][4];
#pragma unroll
  for (int i = 0; i < 2; ++i)
#pragma unroll
    for (int j = 0; j < 4; ++j)
#pragma unroll
      for (int v = 0; v < 8; ++v) acc[i][j][v] = 0.0f;

  const float* a0p = Xb + (long)(m0 + idx) * K;
  const float* a1p = Xb + (long)(m0 + 16 + idx) * K;
  const float* bp0 = Wb + (long)(n0 +  0 + idx) * K;
  const float* bp1 = Wb + (long)(n0 + 16 + idx) * K;
  const float* bp2 = Wb + (long)(n0 + 32 + idx) * K;
  const float* bp3 = Wb + (long)(n0 + 48 + idx) * K;

  for (int k0 = 0; k0 < K; k0 += 4) {
    if ((k0 & 63) == 0) {          // light read-ahead -> global_prefetch_b8
      __builtin_prefetch(a0p + k0 + 64, 0, 1);
      __builtin_prefetch(bp0 + k0 + 64, 0, 1);
    }
    int kk = k0 + 2 * half;
    v2f a0 = *(const v2f*)(a0p + kk);
    v2f a1 = *(const v2f*)(a1p + kk);
    v2f b0 = *(const v2f*)(bp0 + kk);
    v2f b1 = *(const v2f*)(bp1 + kk);
    v2f b2 = *(const v2f*)(bp2 + kk);
    v2f b3 = *(const v2f*)(bp3 + kk);
    acc[0][0] = __builtin_amdgcn_wmma_f32_16x16x4_f32(false, a0, false, b0, (short)0, acc[0][0], false, false);
    acc[0][1] = __builtin_amdgcn_wmma_f32_16x16x4_f32(false, a0, false, b1, (short)0, acc[0][1], false, false);
    acc[0][2] = __builtin_amdgcn_wmma_f32_16x16x4_f32(false, a0, false, b2, (short)0, acc[0][2], false, false);
    acc[0][3] = __builtin_amdgcn_wmma_f32_16x16x4_f32(false, a0, false, b3, (short)0, acc[0][3], false, false);
    acc[1][0] = __builtin_amdgcn_wmma_f32_16x16x4_f32(false, a1, false, b0, (short)0, acc[1][0], false, false);
    acc[1][1] = __builtin_amdgcn_wmma_f32_16x16x4_f32(false, a1, false, b1, (short)0, acc[1][1], false, false);
    acc[1][2] = __builtin_amdgcn_wmma_f32_16x16x4_f32(false, a1, false, b2, (short)0, acc[1][2], false, false);
    acc[1][3] = __builtin_amdgcn_wmma_f32_16x16x4_f32(false, a1, false, b3, (short)0, acc[1][3], false, false);
  }

  float s = scale_const * (scale_ptr ? scale_ptr[0] : 1.0f);
#pragma unroll
  for (int i = 0; i < 2; ++i) {
#pragma unroll
    for (int j = 0; j < 4; ++j) {
      int col = n0 + 16 * j + idx;
      float bv = bias ? bias[col] : 0.0f;
#pragma unroll
      for (int v = 0; v < 8; ++v) {
        int row = m0 + 16 * i + v + 8 * half;
        Yb[(long)row * Nn + col] = acc[i][j][v] * s + bv;
      }
    }
  }
}

// ---------------------------------------------------------------------------
// Host side
// ---------------------------------------------------------------------------

static inline void bn_block(float* buf, const float* g, const float* b, float* stats,
                            int C, int W, int reluFirst, int reluAfter, hipStream_t s) {
  long total = (long)NS * C * W;
  zero_k<<<(2 * C + 255) / 256, 256, 0, s>>>(stats, 2 * C);
  stats_k<<<2048, 256, 0, s>>>(buf, stats, total, C, W, reluFirst);
  bnfin_k<<<(C + 255) / 256, 256, 0, s>>>(stats, g, b, C, 1.0f / (float)((long)NS * W));
  bnapply_k<<<(unsigned)((total + 255) / 256), 256, 0, s>>>(buf, stats, total, C, W, reluAfter);
}

static inline void gemm(const float* X, const float* Wt, const float* bias, float* Y,
                        int M, int Nn, int K, int batches, long sx, long sw, long sy,
                        const float* scale_ptr, float scale_const, hipStream_t s) {
  dim3 grid(Nn / 128, M / 64, batches);
  gemm_wmma_k<<<grid, 128, 0, s>>>(X, Wt, bias, Y, M, Nn, K, sx, sw, sy, scale_ptr, scale_const);
}

extern "C" void kernel_launch(void* const* d_in, const int* in_sizes, int n_in,
                              void* d_out, int out_size, void* d_ws, size_t ws_size,
                              hipStream_t stream) {
  (void)in_sizes; (void)n_in; (void)out_size; (void)ws_size;
  const float* EMG      = (const float*)d_in[0];
  const float* GLOVE    = (const float*)d_in[1];
  const float* ce1_w    = (const float*)d_in[2];
  const float* bn_e1_g  = (const float*)d_in[3];
  const float* bn_e1_b  = (const float*)d_in[4];
  const float* ce2_w    = (const float*)d_in[5];
  const float* bn_e2_g  = (const float*)d_in[6];
  const float* bn_e2_b  = (const float*)d_in[7];
  const float* lc1_w    = (const float*)d_in[8];
  const float* bn_l1_g  = (const float*)d_in[9];
  const float* bn_l1_b  = (const float*)d_in[10];
  const float* lc2_w    = (const float*)d_in[11];
  const float* bn_l2_g  = (const float*)d_in[12];
  const float* bn_l2_b  = (const float*)d_in[13];
  const float* le1_w    = (const float*)d_in[14];
  const float* bn_le1_g = (const float*)d_in[15];
  const float* bn_le1_b = (const float*)d_in[16];
  const float* le2_w    = (const float*)d_in[17];
  const float* bn_le2_g = (const float*)d_in[18];
  const float* bn_le2_b = (const float*)d_in[19];
  const float* last_e_w = (const float*)d_in[20];
  const float* cg1_w    = (const float*)d_in[21];
  const float* cg1_b    = (const float*)d_in[22];
  const float* bn_g1_g  = (const float*)d_in[23];
  const float* bn_g1_b  = (const float*)d_in[24];
  const float* cg2_w    = (const float*)d_in[25];
  const float* cg2_b    = (const float*)d_in[26];
  const float* bn_g2_g  = (const float*)d_in[27];
  const float* bn_g2_b  = (const float*)d_in[28];
  const float* lg1_w    = (const float*)d_in[29];
  const float* lg1_b    = (const float*)d_in[30];
  const float* bn_lg1_g = (const float*)d_in[31];
  const float* bn_lg1_b = (const float*)d_in[32];
  const float* lg2_w    = (const float*)d_in[33];
  const float* lg2_b    = (const float*)d_in[34];
  const float* bn_lg2_g = (const float*)d_in[35];
  const float* bn_lg2_b = (const float*)d_in[36];
  const float* lg3_w    = (const float*)d_in[37];
  const float* lg3_b    = (const float*)d_in[38];
  const float* bn_lg3_g = (const float*)d_in[39];
  const float* bn_lg3_b = (const float*)d_in[40];
  const float* last_g_w = (const float*)d_in[41];
  const float* logit_sc = (const float*)d_in[42];
  float* out = (float*)d_out;

  float* ws    = (float*)d_ws;
  float* buf1  = ws;                               // NS*1408
  float* buf2  = buf1 + (long)NS * 1408;           // NS*1408
  float* buf3  = buf2 + (long)NS * 1408;           // NS*512
  float* ebuf  = buf3 + (long)NS * 512;            // NS*128
  float* gbuf  = ebuf + (long)NS * 128;            // NS*128
  float* stats = gbuf + (long)NS * 128;            // 1024

  int tot;
  // ---------------- EMG net ----------------
  tot = NS * 64 * 16;
  conv3_1toC_k<<<(tot + 255) / 256, 256, 0, stream>>>(EMG, ce1_w, nullptr, buf1, NS, 16, 64);
  bn_block(buf1, bn_e1_g, bn_e1_b, stats, 64, 16, 0, 1, stream);
  conv3_CtoC_k<<<(tot + 255) / 256, 256, 0, stream>>>(buf1, ce2_w, nullptr, buf2, NS, 16, 64, 64);
  bn_block(buf2, bn_e2_g, bn_e2_b, stats, 64, 16, 0, 1, stream);
  tot = NS * 32 * 16;
  lc_k<<<(tot + 255) / 256, 256, 0, stream>>>(buf2, lc1_w, buf3, NS, 16, 64, 32);
  bn_block(buf3, bn_l1_g, bn_l1_b, stats, 32, 16, 0, 1, stream);
  lc_k<<<(tot + 255) / 256, 256, 0, stream>>>(buf3, lc2_w, buf1, NS, 16, 32, 32);
  bn_block(buf1, bn_l2_g, bn_l2_b, stats, 32, 16, 0, 1, stream);
  gemm(buf1, le1_w, nullptr, buf2, NS, 512, 512, 1, 0, 0, 0, nullptr, 1.0f, stream);
  bn_block(buf2, bn_le1_g, bn_le1_b, stats, 512, 1, 0, 1, stream);
  gemm(buf2, le2_w, nullptr, buf3, NS, 512, 512, 1, 0, 0, 0, nullptr, 1.0f, stream);
  bn_block(buf3, bn_le2_g, bn_le2_b, stats, 512, 1, 0, 1, stream);
  gemm(buf3, last_e_w, nullptr, ebuf, NS, 128, 512, 1, 0, 0, 0, nullptr, 1.0f, stream);
  l2norm_k<<<NS, 128, 0, stream>>>(ebuf, 128);

  // ---------------- GLOVE net ----------------
  tot = NS * 64 * 22;
  conv3_1toC_k<<<(tot + 255) / 256, 256, 0, stream>>>(GLOVE, cg1_w, cg1_b, buf1, NS, 22, 64);
  bn_block(buf1, bn_g1_g, bn_g1_b, stats, 64, 22, 1, 0, stream);
  conv3_CtoC_k<<<(tot + 255) / 256, 256, 0, stream>>>(buf1, cg2_w, cg2_b, buf2, NS, 22, 64, 64);
  bn_block(buf2, bn_g2_g, bn_g2_b, stats, 64, 22, 1, 0, stream);
  gemm(buf2, lg1_w, lg1_b, buf3, NS, 256, 1408, 1, 0, 0, 0, nullptr, 1.0f, stream);
  bn_block(buf3, bn_lg1_g, bn_lg1_b, stats, 256, 1, 1, 0, stream);
  gemm(buf3, lg2_w, lg2_b, buf1, NS, 256, 256, 1, 0, 0, 0, nullptr, 1.0f, stream);
  bn_block(buf1, bn_lg2_g, bn_lg2_b, stats, 256, 1, 1, 0, stream);
  gemm(buf1, lg3_w, lg3_b, buf3, NS, 256, 256, 1, 0, 0, 0, nullptr, 1.0f, stream);
  bn_block(buf3, bn_lg3_g, bn_lg3_b, stats, 256, 1, 1, 0, stream);
  gemm(buf3, last_g_w, nullptr, gbuf, NS, 128, 256, 1, 0, 0, 0, nullptr, 1.0f, stream);
  l2norm_k<<<NS, 128, 0, stream>>>(gbuf, 128);

  // ---------------- batched logits: (e*100) @ (g*100)^T * logit_scale ----------------
  gemm(ebuf, gbuf, nullptr, out, 128, 128, 128, 128,
       128L * 128, 128L * 128, 128L * 128, logit_sc, 10000.0f, stream);
}